// ProjGA_72438918414848
// MI455X (gfx1250) — compile-verified
//
#include <hip/hip_runtime.h>

// ProjGA hybrid-beamforming optimizer for MI455X (gfx1250).
// One workgroup (256 thr = 8 wave32) per batch; all per-batch matrices live in
// LDS (~201KB of 320KB/WGP). All GEMMs are complex fp32 via V_WMMA_F32_16X16X4_F32.
// Inputs staged HBM->LDS by the Tensor Data Mover with hardware row padding
// (stride 64 -> 66 floats, bank-conflict-free). inv(M)/slogdet(M) via Cholesky
// (M = I + hw hw^H is Hermitian PD). SVD(h) top-16 right singular vectors via
// orthogonal iteration on G = hs^H hs.

typedef float v2f __attribute__((ext_vector_type(2)));
typedef float v8f __attribute__((ext_vector_type(8)));
typedef unsigned int v4u __attribute__((ext_vector_type(4)));
typedef int v4i __attribute__((ext_vector_type(4)));
typedef int v8i __attribute__((ext_vector_type(8)));

#define NT 256

constexpr int LDN = 66;              // padded leading dim for 64-wide matrices
constexpr int LDK = 18;              // padded leading dim for 16-wide matrices
constexpr int SZ_BIG = 64 * LDN;
constexpr int SZ_WA  = 64 * LDK;
constexpr int SZ_WD  = 16 * LDN;

constexpr int O_HSR = 0;
constexpr int O_HSI = O_HSR + SZ_BIG;
constexpr int O_HWR = O_HSI + SZ_BIG;
constexpr int O_HWI = O_HWR + SZ_BIG;
constexpr int O_MR  = O_HWI + SZ_BIG;
constexpr int O_MI  = O_MR  + SZ_BIG;
constexpr int O_XR  = O_MI  + SZ_BIG;
constexpr int O_XI  = O_XR  + SZ_BIG;
constexpr int O_PR  = O_XI  + SZ_BIG;
constexpr int O_PI  = O_PR  + SZ_BIG;
constexpr int O_WAR = O_PI  + SZ_BIG;
constexpr int O_WAI = O_WAR + SZ_WA;
constexpr int O_WDR = O_WAI + SZ_WA;
constexpr int O_WDI = O_WDR + SZ_WD;
constexpr int O_T1R = O_WDI + SZ_WD;
constexpr int O_T1I = O_T1R + SZ_WA;
constexpr int O_GR  = O_T1I + SZ_WA;
constexpr int O_GI  = O_GR  + SZ_WA;   // grad scratch: max(64x16 ld18, 16x64 ld66)
constexpr int O_RED = O_GI  + SZ_WA;
constexpr int O_LOG = O_RED + NT;
constexpr int SMEM_FLOATS = O_LOG + 4;

// ---------------------------------------------------------------------------
// TDM: DMA `total_elems` fp32 from global into LDS, inserting 2 pad DWORDs
// after every 64 DWORDs (rows of 64 land on stride-66 padded layout).
// D# group0: count=1 | lds_addr | global_addr[56:0] | type=2.
// D# group1: data_size=4B, pad_enable, pad_interval=64DW(code 5),
//            pad_amount=2DW(code 1), tensor_dim0 = tile_dim0 = total_elems (1-D).
// Groups 2/3 zero (<=2D tensor). Tracked by TENSORcnt.
// ---------------------------------------------------------------------------
__device__ __forceinline__ void tdm_load_rows_padded(const void* g,
                                                     unsigned lds_off_bytes,
                                                     unsigned total_elems)
{
  const unsigned long long ga = (unsigned long long)g;
  v4u g0;
  g0.x = 1u;                                                   // count=1 valid
  g0.y = lds_off_bytes;                                        // lds_addr
  g0.z = (unsigned)ga;                                         // global[31:0]
  g0.w = (unsigned)((ga >> 32) & 0x01FFFFFFull) | 0x80000000u; // [56:32]|type=2
  v8i g1 = {};
  g1[0] = (int)((2u << 16)        // data_size = 4 bytes
              | (1u << 20)        // pad_enable
              | (5u << 22)        // pad_interval code 5 = 64 DWORDs
              | (1u << 25));      // pad_amount  code 1 = 2 DWORDs
  g1[1] = (int)((total_elems & 0xFFFFu) << 16);   // tensor_dim0[15:0]
  g1[2] = (int)((total_elems >> 16) & 0xFFFFu);   // tensor_dim0[31:16]
  g1[3] = (int)((total_elems & 0xFFFFu) << 16);   // tile_dim0
  const v4i z4 = {};
#if __clang_major__ >= 23
  const v8i z8 = {};
  __builtin_amdgcn_tensor_load_to_lds(g0, g1, z4, z4, z8, 0);
#else
  __builtin_amdgcn_tensor_load_to_lds(g0, g1, z4, z4, 0);
#endif
}

// ---------------------------------------------------------------------------
// Complex 16x16 output tile of C = op(A) * op(B) using V_WMMA_F32_16X16X4_F32.
// TA/TB: operand stored transposed (access elem(m,k) = A[k*ld+m]); CA/CB: conj.
// ---------------------------------------------------------------------------
template<bool TA, bool CA, bool TB, bool CB>
__device__ __forceinline__ void cgemm_tile(
    const float* __restrict__ AR, const float* __restrict__ AI, int lda,
    const float* __restrict__ BR, const float* __restrict__ BI, int ldb,
    int K, int row0, int col0,
    float* __restrict__ CR, float* __restrict__ CI, int ldc, bool addI)
{
  const int lane = threadIdx.x & 31;
  const int half = lane >> 4;
  const int l15  = lane & 15;
  const int m = row0 + l15;
  const int n = col0 + l15;
  v8f cr = {};
  v8f ci = {};
  for (int k0 = 0; k0 < K; k0 += 4) {
    const int kb = k0 + (half << 1);
    v2f ar, ai, br, bi;
    if (TA) {
      ar.x = AR[(kb + 0) * lda + m]; ar.y = AR[(kb + 1) * lda + m];
      ai.x = AI[(kb + 0) * lda + m]; ai.y = AI[(kb + 1) * lda + m];
    } else {
      ar.x = AR[m * lda + kb + 0]; ar.y = AR[m * lda + kb + 1];
      ai.x = AI[m * lda + kb + 0]; ai.y = AI[m * lda + kb + 1];
    }
    if (CA) { ai.x = -ai.x; ai.y = -ai.y; }
    if (TB) {
      br.x = BR[n * ldb + kb + 0]; br.y = BR[n * ldb + kb + 1];
      bi.x = BI[n * ldb + kb + 0]; bi.y = BI[n * ldb + kb + 1];
    } else {
      br.x = BR[(kb + 0) * ldb + n]; br.y = BR[(kb + 1) * ldb + n];
      bi.x = BI[(kb + 0) * ldb + n]; bi.y = BI[(kb + 1) * ldb + n];
    }
    if (CB) { bi.x = -bi.x; bi.y = -bi.y; }
    v2f nai; nai.x = -ai.x; nai.y = -ai.y;
    // Re += Ar*Br - Ai*Bi ; Im += Ar*Bi + Ai*Br
    cr = __builtin_amdgcn_wmma_f32_16x16x4_f32(false, ar,  false, br, (short)0, cr, false, false);
    cr = __builtin_amdgcn_wmma_f32_16x16x4_f32(false, nai, false, bi, (short)0, cr, false, false);
    ci = __builtin_amdgcn_wmma_f32_16x16x4_f32(false, ar,  false, bi, (short)0, ci, false, false);
    ci = __builtin_amdgcn_wmma_f32_16x16x4_f32(false, ai,  false, br, (short)0, ci, false, false);
  }
#pragma unroll
  for (int r = 0; r < 8; ++r) {
    const int mm = row0 + half * 8 + r;
    const float add = (addI && (mm == n)) ? 1.0f : 0.0f;
    CR[mm * ldc + n] = cr[r] + add;
    CI[mm * ldc + n] = ci[r];
  }
}

// Wave-parallel complex GEMM over 16x16 tiles. Callers barrier around it.
template<bool TA, bool CA, bool TB, bool CB>
__device__ __forceinline__ void cgemm(
    const float* AR, const float* AI, int lda,
    const float* BR, const float* BI, int ldb,
    float* CR, float* CI, int ldc,
    int M, int N, int K, bool addI)
{
  const int wave = threadIdx.x >> 5;
  const int tm = M >> 4, tn = N >> 4;
  for (int t = wave; t < tm * tn; t += (NT / 32)) {
    const int r0 = (t / tn) << 4;
    const int c0 = (t % tn) << 4;
    cgemm_tile<TA, CA, TB, CB>(AR, AI, lda, BR, BI, ldb, K, r0, c0, CR, CI, ldc, addI);
  }
}

__device__ __forceinline__ float block_reduce_sum(float v, float* red, int tid)
{
  red[tid] = v;
  __syncthreads();
  for (int s = NT / 2; s > 0; s >>= 1) {
    if (tid < s) red[tid] += red[tid + s];
    __syncthreads();
  }
  const float r = red[0];
  __syncthreads();
  return r;
}

// In-place Cholesky of Hermitian 64x64 (lower triangle -> L); logdet to sLog[0].
__device__ void cholesky64(float* MR, float* MI, float* red, float* sLog, int tid)
{
  if (tid == 0) sLog[0] = 0.0f;
  __syncthreads();
  for (int k = 0; k < 64; ++k) {
    if (tid == 0) {
      float d = sqrtf(fmaxf(MR[k * LDN + k], 1e-30f));
      MR[k * LDN + k] = d;
      MI[k * LDN + k] = 0.0f;
      sLog[0] += 2.0f * logf(d);
      red[0] = 1.0f / d;
    }
    __syncthreads();
    const float inv = red[0];
    if (tid > k && tid < 64) {
      MR[tid * LDN + k] *= inv;
      MI[tid * LDN + k] *= inv;
    }
    __syncthreads();
    const int rem = 63 - k;
    for (int idx = tid; idx < rem * rem; idx += NT) {
      const int i = k + 1 + idx / rem;
      const int j = k + 1 + idx % rem;
      if (j <= i) {   // M[i][j] -= L[i][k] * conj(L[j][k])
        const float lr = MR[i * LDN + k], li = MI[i * LDN + k];
        const float mr = MR[j * LDN + k], mi = MI[j * LDN + k];
        MR[i * LDN + j] -= lr * mr + li * mi;
        MI[i * LDN + j] -= li * mr - lr * mi;
      }
    }
    __syncthreads();
  }
}

// Solve M X = B in place (X holds RHS on entry) using L: forward + adjoint back-sub.
// One thread per RHS column: zero barriers, lane-contiguous LDS access.
__device__ void solve64(const float* LR, const float* LI, float* XR, float* XI, int tid)
{
  if (tid < 64) {
    const int c = tid;
    for (int i = 0; i < 64; ++i) {
      float ar = XR[i * LDN + c], ai = XI[i * LDN + c];
      for (int k = 0; k < i; ++k) {
        const float lr = LR[i * LDN + k], li = LI[i * LDN + k];
        const float xr = XR[k * LDN + c], xi = XI[k * LDN + c];
        ar -= lr * xr - li * xi;
        ai -= lr * xi + li * xr;
      }
      const float inv = 1.0f / LR[i * LDN + i];
      XR[i * LDN + c] = ar * inv; XI[i * LDN + c] = ai * inv;
    }
    for (int i = 63; i >= 0; --i) {
      float ar = XR[i * LDN + c], ai = XI[i * LDN + c];
      for (int k = i + 1; k < 64; ++k) {
        const float lr = LR[k * LDN + i], li = -LI[k * LDN + i]; // conj(L[k][i])
        const float xr = XR[k * LDN + c], xi = XI[k * LDN + c];
        ar -= lr * xr - li * xi;
        ai -= lr * xi + li * xr;
      }
      const float inv = 1.0f / LR[i * LDN + i];
      XR[i * LDN + c] = ar * inv; XI[i * LDN + c] = ai * inv;
    }
  }
  __syncthreads();
}

// Modified Gram-Schmidt on 64x16 complex V (ld = LDK).
__device__ void mgs64x16(float* VR, float* VI, float* red, int tid)
{
  for (int j = 0; j < 16; ++j) {
    float p = 0.0f;
    if (tid < 64) {
      const float r = VR[tid * LDK + j], im = VI[tid * LDK + j];
      p = r * r + im * im;
    }
    const float nrm2 = block_reduce_sum(p, red, tid);
    const float invn = rsqrtf(nrm2 + 1e-30f);
    if (tid < 64) { VR[tid * LDK + j] *= invn; VI[tid * LDK + j] *= invn; }
    __syncthreads();
    if (tid > j && tid < 16) {    // dot_c = v_j^H v_c
      float dr = 0.0f, di = 0.0f;
      for (int r = 0; r < 64; ++r) {
        const float vr = VR[r * LDK + j], vi = -VI[r * LDK + j];
        const float wr = VR[r * LDK + tid], wi = VI[r * LDK + tid];
        dr += vr * wr - vi * wi;
        di += vr * wi + vi * wr;
      }
      red[2 * tid] = dr; red[2 * tid + 1] = di;
    }
    __syncthreads();
    for (int idx = tid; idx < 64 * 16; idx += NT) {
      const int r = idx >> 4, c = idx & 15;
      if (c > j) {
        const float dr = red[2 * c], di = red[2 * c + 1];
        const float vr = VR[r * LDK + j], vi = VI[r * LDK + j];
        VR[r * LDK + c] -= dr * vr - di * vi;
        VI[r * LDK + c] -= dr * vi + di * vr;
      }
    }
    __syncthreads();
  }
}

// hw = hs*wa*wd ; M = I + hw hw^H ; Cholesky(M) (logdet -> sLog);
// if doSolve: X = M^-1 hw (so conj(X) = M^-T conj(hw)).
__device__ void gram_chol(float* sm, int tid, bool doSolve)
{
  cgemm<false, false, false, false>(sm + O_HSR, sm + O_HSI, LDN,
                                    sm + O_WAR, sm + O_WAI, LDK,
                                    sm + O_T1R, sm + O_T1I, LDK, 64, 16, 64, false);
  __syncthreads();
  cgemm<false, false, false, false>(sm + O_T1R, sm + O_T1I, LDK,
                                    sm + O_WDR, sm + O_WDI, LDN,
                                    sm + O_HWR, sm + O_HWI, LDN, 64, 64, 16, false);
  __syncthreads();
  cgemm<false, false, true, true>(sm + O_HWR, sm + O_HWI, LDN,
                                  sm + O_HWR, sm + O_HWI, LDN,
                                  sm + O_MR, sm + O_MI, LDN, 64, 64, 64, true);
  __syncthreads();
  cholesky64(sm + O_MR, sm + O_MI, sm + O_RED, sm + O_LOG, tid);
  if (doSolve) {
    for (int idx = tid; idx < 4096; idx += NT) {
      const int a = (idx >> 6) * LDN + (idx & 63);
      sm[O_XR + a] = sm[O_HWR + a];
      sm[O_XI + a] = sm[O_HWI + a];
    }
    __syncthreads();
    solve64(sm + O_MR, sm + O_MI, sm + O_XR, sm + O_XI, tid);
  }
}

// P = hs^T * conj(X)   (= hs^T M^-T conj(hw))
__device__ __forceinline__ void compute_P(float* sm)
{
  cgemm<true, false, false, true>(sm + O_HSR, sm + O_HSI, LDN,
                                  sm + O_XR, sm + O_XI, LDN,
                                  sm + O_PR, sm + O_PI, LDN, 64, 64, 64, false);
}

__global__ __launch_bounds__(NT) void projga_kernel(
    const float* __restrict__ hR, const float* __restrict__ hI,
    const float* __restrict__ hyp,
    const float* __restrict__ wd0R, const float* __restrict__ wd0I,
    const float* __restrict__ nvp,
    float* __restrict__ out, int B, int T)
{
  extern __shared__ float sm[];
  const int tid = threadIdx.x;
  const int b = blockIdx.x;

  float* waR = sm + O_WAR; float* waI = sm + O_WAI;
  float* wdR = sm + O_WDR; float* wdI = sm + O_WDI;
  float* gR  = sm + O_GR;  float* gI  = sm + O_GI;
  float* red = sm + O_RED;

  // output sections (floats): obj(B*T) | wa | wd | W_A | W_D (complex interleaved)
  const size_t OFF_WAF = (size_t)B * (size_t)T;
  const size_t OFF_WDF = OFF_WAF + (size_t)B * 2048;
  const size_t OFF_WAH = OFF_WDF + (size_t)B * 2048;
  const size_t OFF_WDH = OFF_WAH + (size_t)T * (size_t)B * 2048;

  // ---- TDM-stage h and wd0 planes into padded LDS (wave 0 issues the DMAs);
  //      hardware pad (64DW interval, 2DW pad) realizes the stride-66 layout.
  if (tid < 32) {
    tdm_load_rows_padded(hR   + (size_t)b * 4096, (unsigned)(O_HSR * 4), 4096u);
    tdm_load_rows_padded(hI   + (size_t)b * 4096, (unsigned)(O_HSI * 4), 4096u);
    tdm_load_rows_padded(wd0R + (size_t)b * 1024, (unsigned)(O_WDR * 4), 1024u);
    tdm_load_rows_padded(wd0I + (size_t)b * 1024, (unsigned)(O_WDI * 4), 1024u);
    __builtin_amdgcn_s_wait_tensorcnt(0);
  }
  __syncthreads();

  // hs = sqrt(1/(n*noise_var)) * h  (scale in place)
  const float scale = sqrtf(1.0f / (64.0f * nvp[0]));
  for (int idx = tid; idx < 4096; idx += NT) {
    const int a = (idx >> 6) * LDN + (idx & 63);
    sm[O_HSR + a] *= scale;
    sm[O_HSI + a] *= scale;
  }
  __syncthreads();

  // ---- top-16 right singular vectors via orthogonal iteration on G=hs^H hs
  cgemm<true, true, false, false>(sm + O_HSR, sm + O_HSI, LDN,
                                  sm + O_HSR, sm + O_HSI, LDN,
                                  sm + O_MR, sm + O_MI, LDN, 64, 64, 64, false);
  __syncthreads();
  for (int idx = tid; idx < 1024; idx += NT) {
    const int r = idx >> 4, c = idx & 15;
    sm[O_T1R + r * LDK + c] = (r == c) ? 1.0f : 0.0f;
    sm[O_T1I + r * LDK + c] = 0.0f;
  }
  __syncthreads();
  for (int it = 0; it < 16; ++it) {
    cgemm<false, false, false, false>(sm + O_MR, sm + O_MI, LDN,
                                      sm + O_T1R, sm + O_T1I, LDK,
                                      gR, gI, LDK, 64, 16, 64, false);
    __syncthreads();
    for (int idx = tid; idx < 1024; idx += NT) {
      const int a = (idx >> 4) * LDK + (idx & 15);
      sm[O_T1R + a] = gR[a];
      sm[O_T1I + a] = gI[a];
    }
    __syncthreads();
    mgs64x16(sm + O_T1R, sm + O_T1I, red, tid);
  }
  // wa = v / |v + EPS|  (phase projection)
  for (int idx = tid; idx < 1024; idx += NT) {
    const int a = (idx >> 4) * LDK + (idx & 15);
    const float re = sm[O_T1R + a], im = sm[O_T1I + a];
    const float mag = fmaxf(sqrtf((re + 1e-12f) * (re + 1e-12f) + im * im), 1e-30f);
    waR[a] = re / mag; waI[a] = im / mag;
  }
  __syncthreads();

  // ---- wd Frobenius normalization (wd0 already staged by TDM) -------------
  {
    cgemm<false, false, false, false>(waR, waI, LDK, wdR, wdI, LDN,
                                      sm + O_XR, sm + O_XI, LDN, 64, 64, 16, false);
    __syncthreads();
    float p = 0.0f;
    for (int idx = tid; idx < 4096; idx += NT) {
      const int a = (idx >> 6) * LDN + (idx & 63);
      p += sm[O_XR + a] * sm[O_XR + a] + sm[O_XI + a] * sm[O_XI + a];
    }
    const float f2 = block_reduce_sum(p, red, tid);
    const float s = sqrtf(64.0f / fmaxf(f2, 1e-30f));
    for (int idx = tid; idx < 1024; idx += NT) {
      const int a = (idx >> 6) * LDN + (idx & 63);
      wdR[a] *= s; wdI[a] *= s;
    }
    __syncthreads();
  }

  // ---- projected gradient ascent loop -------------------------------------
  for (int t = 0; t < T; ++t) {
    const float h0 = hyp[2 * t + 0];
    const float h1 = hyp[2 * t + 1];

    // grad_wa = hs^T M^-T conj(hw) wd^T = P wd^T
    gram_chol(sm, tid, true);
    compute_P(sm);
    __syncthreads();
    cgemm<false, false, true, false>(sm + O_PR, sm + O_PI, LDN,
                                     wdR, wdI, LDN,
                                     gR, gI, LDK, 64, 16, 64, false);
    __syncthreads();
    // wa <- phase(wa + h0 * grad_wa)
    for (int idx = tid; idx < 1024; idx += NT) {
      const int a = (idx >> 4) * LDK + (idx & 15);
      const float re = waR[a] + h0 * gR[a];
      const float im = waI[a] + h0 * gI[a];
      const float mag = fmaxf(sqrtf((re + 1e-12f) * (re + 1e-12f) + im * im), 1e-30f);
      waR[a] = re / mag; waI[a] = im / mag;
    }
    __syncthreads();

    // grad_wd = wa^T P  (with gram recomputed at the NEW wa)
    gram_chol(sm, tid, true);
    compute_P(sm);
    __syncthreads();
    cgemm<true, false, false, false>(waR, waI, LDK,
                                     sm + O_PR, sm + O_PI, LDN,
                                     gR, gI, LDN, 16, 64, 64, false);
    __syncthreads();
    // wd <- normalize(wd + h1 * grad_wd)
    for (int idx = tid; idx < 1024; idx += NT) {
      const int a = (idx >> 6) * LDN + (idx & 63);
      wdR[a] += h1 * gR[a];
      wdI[a] += h1 * gI[a];
    }
    __syncthreads();
    cgemm<false, false, false, false>(waR, waI, LDK, wdR, wdI, LDN,
                                      sm + O_XR, sm + O_XI, LDN, 64, 64, 16, false);
    __syncthreads();
    float p = 0.0f;
    for (int idx = tid; idx < 4096; idx += NT) {
      const int a = (idx >> 6) * LDN + (idx & 63);
      p += sm[O_XR + a] * sm[O_XR + a] + sm[O_XI + a] * sm[O_XI + a];
    }
    const float f2 = block_reduce_sum(p, red, tid);
    const float s = sqrtf(64.0f / fmaxf(f2, 1e-30f));
    for (int idx = tid; idx < 1024; idx += NT) {
      const int a = (idx >> 6) * LDN + (idx & 63);
      wdR[a] *= s; wdI[a] *= s;
    }
    __syncthreads();

    // objective: |logdet(I + hw hw^H)| at (wa, wd)  — Cholesky only
    gram_chol(sm, tid, false);

    // ---- stream outputs ---------------------------------------------------
    for (int idx = tid; idx < 1024; idx += NT) {
      const int ra = (idx >> 4) * LDK + (idx & 15);   // wa layout 64x16
      const int rd = (idx >> 6) * LDN + (idx & 63);   // wd layout 16x64
      const size_t oa = OFF_WAH + (((size_t)t * B + b) * 1024 + idx) * 2;
      out[oa + 0] = waR[ra]; out[oa + 1] = waI[ra];
      const size_t od = OFF_WDH + (((size_t)t * B + b) * 1024 + idx) * 2;
      out[od + 0] = wdR[rd]; out[od + 1] = wdI[rd];
      if (t == T - 1) {
        const size_t fa = OFF_WAF + ((size_t)b * 1024 + idx) * 2;
        out[fa + 0] = waR[ra]; out[fa + 1] = waI[ra];
        const size_t fd = OFF_WDF + ((size_t)b * 1024 + idx) * 2;
        out[fd + 0] = wdR[rd]; out[fd + 1] = wdI[rd];
      }
    }
    if (tid == 0) out[(size_t)b * T + t] = fabsf(sm[O_LOG]);
    __syncthreads();
  }
}

extern "C" void kernel_launch(void* const* d_in, const int* in_sizes, int n_in,
                              void* d_out, int out_size, void* d_ws, size_t ws_size,
                              hipStream_t stream)
{
  (void)n_in; (void)out_size; (void)d_ws; (void)ws_size;
  const float* hR   = (const float*)d_in[0];
  const float* hI   = (const float*)d_in[1];
  const float* hyp  = (const float*)d_in[2];
  const float* wd0R = (const float*)d_in[3];
  const float* wd0I = (const float*)d_in[4];
  const float* nv   = (const float*)d_in[5];
  float* out = (float*)d_out;

  const int B = in_sizes[0] / (64 * 64);   // 4096
  const int T = in_sizes[2] / 2;           // 20

  const size_t smem = (size_t)SMEM_FLOATS * sizeof(float);
  hipLaunchKernelGGL(projga_kernel, dim3(B), dim3(NT), smem, stream,
                     hR, hI, hyp, wd0R, wd0I, nv, out, B, T);
}